// inverse_LDS_60619168415865
// MI455X (gfx1250) — compile-verified
//
#include <hip/hip_runtime.h>
#include <hip/hip_bf16.h>
#include <math.h>

// ---------------------------------------------------------------------------
// inverse_LDS for MI455X (gfx1250, wave32, WMMA)
//
// Phases:
//  1) recog_gemm<N,DIAG> : data[T,64] @ W[64,N] via v_wmma_f32_16x16x4_f32,
//                          N compile-time so B-frag loads use immediate offsets
//  2) pair_products      : per-t 16x16x16 A*A^T, B*B^T, A*B^T via wmma
//  3) chol_scan          : serial block-tridiag Cholesky, 1 persistent wave,
//                          Y^T*Y Schur update via wmma, entropy accumulation
//  4) back_scan          : serial backward block substitution, 1 wave
// ---------------------------------------------------------------------------

#define T_LEN 65536
#define ZD    16
#define XD    64

typedef __attribute__((ext_vector_type(2))) float v2f;
typedef __attribute__((ext_vector_type(8))) float v8f;

static __device__ __forceinline__ v8f wmma_f32_k4(v2f a, v2f b, v8f c) {
    // D = A(16x4,f32) * B(4x16,f32) + C(16x16,f32)
    return __builtin_amdgcn_wmma_f32_16x16x4_f32(
        /*neg_a=*/false, a, /*neg_b=*/false, b,
        /*c_mod=*/(short)0, c, /*reuse_a=*/false, /*reuse_b=*/false);
}

// ---------------------------------------------------------------------------
// Phase 1: out[T, N] = data[T,64] @ W[64,N] + bias   (N is compile-time)
// ADD_DIAG: add 3.0 on the per-t matrix diagonal (column 16*nt + nt).
// One wave per (16-row tile, 16-col tile). K=64 -> 16 chained wmma.
// A-frag:  a.x = data[r0+l15][4kb+2hi], a.y = +1      (one b64 load)
// B-frag:  b.x = W[(4kb+2hi)*N + n],    b.y = +N row  (const-offset b32 loads)
// ---------------------------------------------------------------------------
template <int N, bool ADD_DIAG>
__global__ __launch_bounds__(32)
void recog_gemm(const float* __restrict__ data,
                const float* __restrict__ W,
                const float* __restrict__ bias,
                float* __restrict__ outp) {
    const int lane = threadIdx.x & 31;
    const int l15  = lane & 15;      // M-row for A-frag, N-col for B/C-frag
    const int hi   = lane >> 4;
    const size_t r0 = (size_t)blockIdx.x * 16;   // first time step of tile
    const int nt = blockIdx.y;                   // 16-column tile
    const int n  = nt * 16 + l15;

    // A fragments (data tile rows r0..r0+15)
    const float* pa = data + (r0 + l15) * XD + 2 * hi;
    v2f afrag[16];
#pragma unroll
    for (int kb = 0; kb < 16; ++kb)
        afrag[kb] = *(const v2f*)(pa + 4 * kb);

    // B base pointer: pw[(4kb)*N] / pw[(4kb+1)*N] are compile-time offsets
    const float* pw = W + (size_t)(2 * hi) * N + n;

    v8f acc = {};
#pragma unroll
    for (int kb = 0; kb < 16; ++kb) {
        v2f bfrag;
        bfrag.x = pw[(4 * kb) * N];
        bfrag.y = pw[(4 * kb + 1) * N];
        acc = wmma_f32_k4(afrag[kb], bfrag, acc);
    }

    // epilogue: bias (+3 on per-t matrix diagonal if requested)
    float add = bias[n];
    if (ADD_DIAG && l15 == nt) add += 3.0f;
#pragma unroll
    for (int v = 0; v < 8; ++v) {
        const int m = v + hi * 8;                 // C-frag row
        outp[(r0 + m) * (size_t)N + n] = acc[v] + add;
    }
}

// ---------------------------------------------------------------------------
// Phase 2: one wave per time step t.
//   AA[t] = AAChol[t] AAChol[t]^T + (t>0 ? BBfull[t] BBfull[t]^T : 0) + 1e-6 I
//   BB[t] = AAChol[t] BBfull[t+1]^T                      (t < T-1)
// For M @ N^T both wmma operand fragments use the same indexing pattern:
//   frag(X, kb).x = X[l15*16 + 4kb+2hi], .y = +1
// NOTE: AAout aliases AAChol (each wave reads its 256 floats before writing).
// ---------------------------------------------------------------------------
static __device__ __forceinline__ v2f frag16(const float* m, int l15, int hi, int kb) {
    return *(const v2f*)(m + l15 * 16 + 4 * kb + 2 * hi);
}

__global__ __launch_bounds__(256)
void pair_products(const float* __restrict__ AAChol,
                   const float* __restrict__ BBfull,
                   float* __restrict__ AAout,      // may alias AAChol
                   float* __restrict__ BBout) {
    const int lane = threadIdx.x & 31;
    const int l15  = lane & 15;
    const int hi   = lane >> 4;
    const size_t t = (size_t)blockIdx.x * 8 + (threadIdx.x >> 5);

    const float* At = AAChol + t * 256;
    v2f f[4];
#pragma unroll
    for (int kb = 0; kb < 4; ++kb) f[kb] = frag16(At, l15, hi, kb);

    // diagsquare (+ odsquare for t>0)
    v8f acc = {};
#pragma unroll
    for (int kb = 0; kb < 4; ++kb) acc = wmma_f32_k4(f[kb], f[kb], acc);
    if (t > 0) {
        const float* Gt = BBfull + t * 256;
#pragma unroll
        for (int kb = 0; kb < 4; ++kb) {
            v2f q = frag16(Gt, l15, hi, kb);
            acc = wmma_f32_k4(q, q, acc);
        }
    }

    // BB[t] = AAChol[t] @ BBfull[t+1]^T
    v8f acc2 = {};
    const bool hasBB = (t + 1 < T_LEN);
    if (hasBB) {
        const float* Nt = BBfull + (t + 1) * 256;
#pragma unroll
        for (int kb = 0; kb < 4; ++kb)
            acc2 = wmma_f32_k4(f[kb], frag16(Nt, l15, hi, kb), acc2);
    }

    // stores (after all reads of AAChol[t] -> in-place alias is safe)
#pragma unroll
    for (int v = 0; v < 8; ++v) {
        const int m = v + hi * 8;
        AAout[t * 256 + m * 16 + l15] = acc[v] + ((m == l15) ? 1e-6f : 0.0f);
    }
    if (hasBB) {
#pragma unroll
        for (int v = 0; v < 8; ++v)
            BBout[t * 256 + (v + hi * 8) * 16 + l15] = acc2[v];
    }
}

// ---------------------------------------------------------------------------
// Phase 3: serial scan, one wave.
//   t=0:  L0 = chol(AA[0])
//   t>0:  Y = Lprev^{-1} BB[t-1]  (forward substitution, Lprev in LDS)
//         C[t-1] = Y^T            (stored; aliases BB buffer in place)
//         G = Y^T Y via wmma (LDS-staged fragments)
//         L[t] = chol(AA[t] - G)  (lanes-as-columns, shfl broadcasts)
//   entropy = -sum_t sum_k log(L[t][k][k])
//   Lout aliases AA buffer (L[t] written after AA[t] consumed).
// ---------------------------------------------------------------------------
__global__ __launch_bounds__(32)
void chol_scan(const float* __restrict__ AAbuf,
               const float* __restrict__ BBbuf,
               float* __restrict__ Lout,        // may alias AAbuf
               float* __restrict__ Cout,        // may alias BBbuf
               float* __restrict__ entropy_out) {
    __shared__ float ldsL[256];
    __shared__ float ldsY[256];
    __shared__ float ldsM[256];
    const int lane = threadIdx.x & 31;
    const int l15  = lane & 15;                 // column index this lane owns
    const int hi   = lane >> 4;
    float logsum = 0.0f;

#pragma unroll 1
    for (int t = 0; t < T_LEN; ++t) {
        v8f g = {};                             // G = Y^T Y (C-layout)
        if (t > 0) {
            // forward substitution: Lprev * Y = BB[t-1]; lane holds column l15
            const float* Bt = BBbuf + (size_t)(t - 1) * 256;
            float y[16];
#pragma unroll
            for (int i = 0; i < 16; ++i) {
                float s = Bt[i * 16 + l15];
#pragma unroll
                for (int j = 0; j < i; ++j) s -= ldsL[i * 16 + j] * y[j];
                y[i] = s / ldsL[i * 16 + i];
            }
            // C[t-1] = Y^T : lane l15 owns row l15 of C; stage Y in LDS for wmma
            if (lane < 16) {
                float* Cp = Cout + (size_t)(t - 1) * 256 + l15 * 16;
#pragma unroll
                for (int j = 0; j < 16; ++j) { Cp[j] = y[j]; ldsY[j * 16 + l15] = y[j]; }
            }
            // G = Y^T Y : A-frag of Y^T == B-frag of Y == ldsY[k][col]
#pragma unroll
            for (int kb = 0; kb < 4; ++kb) {
                const int k0 = 4 * kb + 2 * hi;
                v2f fr;
                fr.x = ldsY[k0 * 16 + l15];
                fr.y = ldsY[(k0 + 1) * 16 + l15];
                g = wmma_f32_k4(fr, fr, g);
            }
        }

        // M = AA[t] - G, scatter C-layout -> LDS, gather columns
        const float* At = AAbuf + (size_t)t * 256;
#pragma unroll
        for (int v = 0; v < 8; ++v) {
            const int mm = v + hi * 8;
            ldsM[mm * 16 + l15] = At[mm * 16 + l15] - g[v];
        }
        float m[16];
#pragma unroll
        for (int i = 0; i < 16; ++i) m[i] = ldsM[i * 16 + l15];

        // right-looking Cholesky, lanes-as-columns
        float lcol[16];
#pragma unroll
        for (int i = 0; i < 16; ++i) lcol[i] = 0.0f;
#pragma unroll
        for (int k = 0; k < 16; ++k) {
            float ck[16];
#pragma unroll
            for (int i = 0; i < 16; ++i) ck[i] = __shfl(m[i], k, 32);
            const float d    = sqrtf(ck[k]);
            const float invd = 1.0f / d;
            const float ljk  = m[k] * invd;     // L[col][k] on lanes col>k
#pragma unroll
            for (int i = k; i < 16; ++i) {
                const float lik = ck[i] * invd;
                lcol[i] = (l15 == k) ? lik : lcol[i];
                if (i > k) m[i] = (l15 > k) ? (m[i] - lik * ljk) : m[i];
            }
            logsum += __logf(d);                // identical on every lane
        }

        // persist L[t] for next step (LDS) and for back_scan (global)
        if (lane < 16) {
#pragma unroll
            for (int i = 0; i < 16; ++i) {
                const float v = lcol[i];
                ldsL[i * 16 + l15]                   = v;
                Lout[(size_t)t * 256 + i * 16 + l15] = v;
            }
        }
    }
    if (lane == 0) entropy_out[0] = -logsum;    // BETA*0.5*(-2*sum(log diag))
}

// ---------------------------------------------------------------------------
// Phase 4: backward block substitution, one wave; lane i owns x[i].
//   x_{T-1} = solve(L_{T-1}^T, b_{T-1})
//   x_t     = solve(L_t^T, b_t - C_t^T x_{t+1})
//   sample[t] = postX[t] + x_t
// ---------------------------------------------------------------------------
static __device__ __forceinline__ float solve_upperT(const float* __restrict__ Lt,
                                                     float y, int c) {
    const float diagL = Lt[c * 16 + c];
    float acc = 0.0f, x = 0.0f;
#pragma unroll
    for (int j = 15; j >= 0; --j) {
        const float xc = (y - acc) / diagL;     // final on lane j at step j
        const float xj = __shfl(xc, j, 32);
        x = (c == j) ? xj : x;
        const float Ljc = Lt[j * 16 + c];       // L^T[c][j] = L[j][c]
        acc += (c < j) ? Ljc * xj : 0.0f;
    }
    return x;
}

__global__ __launch_bounds__(32)
void back_scan(const float* __restrict__ Lbuf,
               const float* __restrict__ Cbuf,
               const float* __restrict__ postX,
               const float* __restrict__ normSamps,
               float* __restrict__ out /* d_out: [0]=entropy, [1..]=sample */) {
    const int lane = threadIdx.x & 31;
    const int c = lane & 15;

    float x;
    {
        const size_t t = T_LEN - 1;
        const float y = normSamps[t * 16 + c];
        x = solve_upperT(Lbuf + t * 256, y, c);
        if (lane < 16) out[1 + t * 16 + c] = postX[t * 16 + c] + x;
    }
#pragma unroll 1
    for (int t = T_LEN - 2; t >= 0; --t) {
        const float* Ct = Cbuf + (size_t)t * 256;
        float y = normSamps[(size_t)t * 16 + c];
#pragma unroll
        for (int k = 0; k < 16; ++k) {
            const float xk = __shfl(x, k, 32);
            y -= Ct[k * 16 + c] * xk;           // (C^T x)[c] = sum_k C[k][c] x[k]
        }
        x = solve_upperT(Lbuf + (size_t)t * 256, y, c);
        if (lane < 16) out[1 + (size_t)t * 16 + c] = postX[(size_t)t * 16 + c] + x;
    }
}

// ---------------------------------------------------------------------------
extern "C" void kernel_launch(void* const* d_in, const int* in_sizes, int n_in,
                              void* d_out, int out_size, void* d_ws, size_t ws_size,
                              hipStream_t stream) {
    (void)in_sizes; (void)n_in; (void)out_size; (void)ws_size;
    const float* data      = (const float*)d_in[0];
    const float* Wmu       = (const float*)d_in[1];
    const float* bmu       = (const float*)d_in[2];
    const float* WL        = (const float*)d_in[3];
    const float* bL        = (const float*)d_in[4];
    const float* WLx       = (const float*)d_in[5];
    const float* bLx       = (const float*)d_in[6];
    const float* normSamps = (const float*)d_in[7];
    float* out = (float*)d_out;

    // workspace layout (floats); heavy aliasing keeps it at ~206 MB:
    //   bufA: AAChol -> AA -> L     bufB: BBfull     bufC: BB -> C
    const size_t T16  = (size_t)T_LEN * 16;
    const size_t T256 = (size_t)T_LEN * 256;
    float* ws    = (float*)d_ws;
    float* postX = ws;
    float* bufA  = postX + T16;
    float* bufB  = bufA + T256;
    float* bufC  = bufB + T256;

    // Phase 1: three statically-specialized GEMMs, 1 wave per 16x16 tile
    recog_gemm<16,  false><<<dim3(T_LEN / 16, 1),  32, 0, stream>>>(data, Wmu, bmu, postX);
    recog_gemm<256, true ><<<dim3(T_LEN / 16, 16), 32, 0, stream>>>(data, WL,  bL,  bufA);
    recog_gemm<256, false><<<dim3(T_LEN / 16, 16), 32, 0, stream>>>(data, WLx, bLx, bufB);

    // Phase 2: 8 waves per block, 1 time step per wave
    pair_products<<<T_LEN / 8, 256, 0, stream>>>(bufA, bufB, bufA, bufC);

    // Phase 3: serial Cholesky scan (writes entropy to out[0])
    chol_scan<<<1, 32, 0, stream>>>(bufA, bufC, bufA, bufC, out);

    // Phase 4: serial backward substitution (writes out[1..])
    back_scan<<<1, 32, 0, stream>>>(bufA, bufC, postX, normSamps, out);
}